// GatedExpert_82248623718986
// MI455X (gfx1250) — compile-verified
//
#include <hip/hip_runtime.h>
#include <hip/hip_bf16.h>

typedef __attribute__((ext_vector_type(2))) float v2f;
typedef __attribute__((ext_vector_type(8))) float v8f;

#define E_    16
#define B_    32768
#define LAT_  128
#define HID_  64
#define CLS_  10

// LDS layout (units: floats). Weights stored TRANSPOSED (n-major) so a WMMA
// B fragment (W[k][n], W[k+1][n]) is one contiguous b64 LDS read. K-strides
// are chosen == 4 (mod 64): bank = 4*laneM + 2*laneHi + {0,1} + const ->
// all 64 banks distinct across a wave's b64 reads (conflict-free).
#define W1T_STRIDE 132              // 128 k + 4 pad
#define W2T_STRIDE 68               // 64 k + 4 pad
#define W3T_STRIDE 68
#define H_STRIDE   68               // h tiles k-major, same bank math
#define H_WAVE     (16 * H_STRIDE)               // 1088 floats per wave slice
#define REGA_F     (8 * H_WAVE)                  // 8704 >= W1T (64*132 = 8448)
#define OFF_W2 (REGA_F)                          // 8704
#define OFF_W3 (OFF_W2 + HID_ * W2T_STRIDE)      // 13056
#define OFF_B1 (OFF_W3 + 16 * W3T_STRIDE)        // 14144
#define OFF_B2 (OFF_B1 + HID_)                   // 14208
#define OFF_B3 (OFF_B2 + HID_)                   // 14272
#define SMEM_F (OFF_B3 + 16)                     // 14288 floats = 57152 B

__global__ void moe_zero_counts(int* counts) {
    if (threadIdx.x < E_) counts[threadIdx.x] = 0;
}

// One-hot routing: find each token's expert, compact token ids per expert.
__global__ void moe_assign(const unsigned char* __restrict__ mask,
                           int* __restrict__ counts, int* __restrict__ perm) {
    int t = blockIdx.x * blockDim.x + threadIdx.x;
    if (t >= B_) return;
    int e = 0;
#pragma unroll
    for (int i = 0; i < E_; ++i)
        e = mask[(size_t)i * B_ + t] ? i : e;   // exactly one is set
    int pos = atomicAdd(&counts[e], 1);
    perm[e * B_ + pos] = t;
}

__launch_bounds__(256)
__global__ void moe_mlp(const float* __restrict__ latent,
                        const float* __restrict__ W1, const float* __restrict__ b1,
                        const float* __restrict__ W2, const float* __restrict__ b2,
                        const float* __restrict__ W3, const float* __restrict__ b3,
                        const int* __restrict__ counts, const int* __restrict__ perm,
                        float* __restrict__ out) {
    __shared__ float smem[SMEM_F];

    const int e    = blockIdx.y;
    const int tile = blockIdx.x;
    const int cnt  = counts[e];
    if (tile * 128 >= cnt) return;            // block-uniform early exit

    const int tid = threadIdx.x;

    // ---- stage expert weights into LDS, transposed (coalesced global reads) ----
    for (int i = tid; i < LAT_ * HID_; i += 256) {
        int k = i >> 6, n = i & 63;
        smem[n * W1T_STRIDE + k] = W1[(size_t)e * LAT_ * HID_ + i];
    }
    for (int i = tid; i < HID_ * HID_; i += 256) {
        int k = i >> 6, n = i & 63;
        smem[OFF_W2 + n * W2T_STRIDE + k] = W2[(size_t)e * HID_ * HID_ + i];
    }
    for (int i = tid; i < HID_ * 16; i += 256) {
        int k = i >> 4, n = i & 15;
        smem[OFF_W3 + n * W3T_STRIDE + k] =
            (n < CLS_) ? W3[(size_t)e * HID_ * CLS_ + k * CLS_ + n] : 0.f;
    }
    if (tid < HID_)                smem[OFF_B1 + tid]        = b1[e * HID_ + tid];
    else if (tid < 2 * HID_)       smem[OFF_B2 + tid - HID_] = b2[e * HID_ + tid - HID_];
    else if (tid < 2 * HID_ + 16) {
        int n = tid - 2 * HID_;
        smem[OFF_B3 + n] = (n < CLS_) ? b3[e * CLS_ + n] : 0.f;
    }
    __syncthreads();

    const int lane    = tid & 31;
    const int wave    = tid >> 5;
    const int laneM   = lane & 15;
    const int laneHi  = lane >> 4;          // 0: K+=0/1, 1: K+=2/3 (f32 A/B layout)
    const int slotBase = tile * 128 + wave * 16;

    const int mySlot = slotBase + laneM;
    const int myTok  = perm[e * B_ + ((mySlot < cnt) ? mySlot : 0)];
    const float* __restrict__ rowp =
        latent + ((size_t)e * B_ + myTok) * LAT_ + 2 * laneHi;

    // Per-lane base pointers for transposed-weight B fragments (column laneM).
    const float* __restrict__ w1p = smem + laneM * W1T_STRIDE + 2 * laneHi;
    const float* __restrict__ w2p = smem + OFF_W2 + laneM * W2T_STRIDE + 2 * laneHi;
    const float* __restrict__ w3p = smem + OFF_W3 + laneM * W3T_STRIDE + 2 * laneHi;

    // ================= layer 1: [16x128] x [128x64], relu =================
    v8f c0 = {}, c1 = {}, c2 = {}, c3 = {};
    for (int k0 = 0; k0 < LAT_; k0 += 4) {
        v2f a  = *(const v2f*)(rowp + k0);   // A: lane=M, VGPR0/1 = K, K+1
        v2f b0 = *(const v2f*)(w1p + k0);                    // cols  0..15
        v2f b1v = *(const v2f*)(w1p + 16 * W1T_STRIDE + k0); // cols 16..31
        v2f b2v = *(const v2f*)(w1p + 32 * W1T_STRIDE + k0); // cols 32..47
        v2f b3v = *(const v2f*)(w1p + 48 * W1T_STRIDE + k0); // cols 48..63
        c0 = __builtin_amdgcn_wmma_f32_16x16x4_f32(false, a, false, b0,  (short)0, c0, false, false);
        c1 = __builtin_amdgcn_wmma_f32_16x16x4_f32(false, a, false, b1v, (short)0, c1, false, false);
        c2 = __builtin_amdgcn_wmma_f32_16x16x4_f32(false, a, false, b2v, (short)0, c2, false, false);
        c3 = __builtin_amdgcn_wmma_f32_16x16x4_f32(false, a, false, b3v, (short)0, c3, false, false);
    }
    {
        const float bb0 = smem[OFF_B1 +  0 + laneM];
        const float bb1 = smem[OFF_B1 + 16 + laneM];
        const float bb2 = smem[OFF_B1 + 32 + laneM];
        const float bb3v = smem[OFF_B1 + 48 + laneM];
#pragma unroll
        for (int v = 0; v < 8; ++v) {
            c0[v] = fmaxf(c0[v] + bb0, 0.f);
            c1[v] = fmaxf(c1[v] + bb1, 0.f);
            c2[v] = fmaxf(c2[v] + bb2, 0.f);
            c3[v] = fmaxf(c3[v] + bb3v, 0.f);
        }
    }

    // W1T is dead from here; its LDS region is reused for wave-private h tiles.
    __syncthreads();
    float* hme = smem + wave * H_WAVE;
#pragma unroll
    for (int v = 0; v < 8; ++v) {
        int m = v + 8 * laneHi;              // C/D layout: VGPR v -> row v / v+8
        hme[m * H_STRIDE +  0 + laneM] = c0[v];
        hme[m * H_STRIDE + 16 + laneM] = c1[v];
        hme[m * H_STRIDE + 32 + laneM] = c2[v];
        hme[m * H_STRIDE + 48 + laneM] = c3[v];
    }

    const float* __restrict__ hp = hme + laneM * H_STRIDE + 2 * laneHi;

    // ================= layer 2: [16x64] x [64x64], relu =================
    v8f d0 = {}, d1 = {}, d2 = {}, d3 = {};
    for (int k0 = 0; k0 < HID_; k0 += 4) {
        v2f a  = *(const v2f*)(hp + k0);
        v2f b0 = *(const v2f*)(w2p + k0);
        v2f b1v = *(const v2f*)(w2p + 16 * W2T_STRIDE + k0);
        v2f b2v = *(const v2f*)(w2p + 32 * W2T_STRIDE + k0);
        v2f b3v = *(const v2f*)(w2p + 48 * W2T_STRIDE + k0);
        d0 = __builtin_amdgcn_wmma_f32_16x16x4_f32(false, a, false, b0,  (short)0, d0, false, false);
        d1 = __builtin_amdgcn_wmma_f32_16x16x4_f32(false, a, false, b1v, (short)0, d1, false, false);
        d2 = __builtin_amdgcn_wmma_f32_16x16x4_f32(false, a, false, b2v, (short)0, d2, false, false);
        d3 = __builtin_amdgcn_wmma_f32_16x16x4_f32(false, a, false, b3v, (short)0, d3, false, false);
    }
    {
        const float bb0 = smem[OFF_B2 +  0 + laneM];
        const float bb1 = smem[OFF_B2 + 16 + laneM];
        const float bb2 = smem[OFF_B2 + 32 + laneM];
        const float bb3v = smem[OFF_B2 + 48 + laneM];
#pragma unroll
        for (int v = 0; v < 8; ++v) {
            d0[v] = fmaxf(d0[v] + bb0, 0.f);
            d1[v] = fmaxf(d1[v] + bb1, 0.f);
            d2[v] = fmaxf(d2[v] + bb2, 0.f);
            d3[v] = fmaxf(d3[v] + bb3v, 0.f);
        }
    }
    // Overwrite this wave's private slice with h2 (in-wave LDS ops are ordered).
#pragma unroll
    for (int v = 0; v < 8; ++v) {
        int m = v + 8 * laneHi;
        hme[m * H_STRIDE +  0 + laneM] = d0[v];
        hme[m * H_STRIDE + 16 + laneM] = d1[v];
        hme[m * H_STRIDE + 32 + laneM] = d2[v];
        hme[m * H_STRIDE + 48 + laneM] = d3[v];
    }

    // ================= layer 3: [16x64] x [64x16(pad of 10)] =================
    v8f f0 = {};
    for (int k0 = 0; k0 < HID_; k0 += 4) {
        v2f a = *(const v2f*)(hp + k0);
        v2f b = *(const v2f*)(w3p + k0);
        f0 = __builtin_amdgcn_wmma_f32_16x16x4_f32(false, a, false, b, (short)0, f0, false, false);
    }
    const float bb3 = smem[OFF_B3 + laneM];

    // Scatter: lane = class column N, VGPR v -> token row M = v + 8*laneHi.
    if (laneM < CLS_) {
#pragma unroll
        for (int v = 0; v < 8; ++v) {
            int slot = slotBase + v + 8 * laneHi;
            if (slot < cnt) {
                int tok = perm[e * B_ + slot];
                out[(size_t)tok * CLS_ + laneM] = f0[v] + bb3;
            }
        }
    }
}

extern "C" void kernel_launch(void* const* d_in, const int* in_sizes, int n_in,
                              void* d_out, int out_size, void* d_ws, size_t ws_size,
                              hipStream_t stream) {
    // setup_inputs order: x, mask, latent, W1, b1, W2, b2, W3, b3
    const unsigned char* mask = (const unsigned char*)d_in[1];  // jnp bool = 1 byte
    const float* latent = (const float*)d_in[2];
    const float* W1 = (const float*)d_in[3];
    const float* b1 = (const float*)d_in[4];
    const float* W2 = (const float*)d_in[5];
    const float* b2 = (const float*)d_in[6];
    const float* W3 = (const float*)d_in[7];
    const float* b3 = (const float*)d_in[8];
    float* out = (float*)d_out;

    int* counts = (int*)d_ws;          // E_ ints
    int* perm   = (int*)d_ws + 16;     // E_*B_ ints (64-byte aligned)

    moe_zero_counts<<<1, 32, 0, stream>>>(counts);
    moe_assign<<<B_ / 256, 256, 0, stream>>>(mask, counts, perm);

    dim3 grid(256, E_);                // 128 tokens per tile, worst case one hot expert
    moe_mlp<<<grid, 256, 0, stream>>>(latent, W1, b1, W2, b2, W3, b3,
                                      counts, perm, out);
}